// ProposalTargetLayer_89069031785249
// MI455X (gfx1250) — compile-verified
//
#include <hip/hip_runtime.h>
#include <math.h>

#define NUM_CLASSES 81
#define ROIS_PER_IMAGE 256
#define FG_ROIS 128
#define FG_THRESH 0.5f
#define BG_HI 0.5f
#define BG_LO 0.1f
#define CAP 256            // we only ever need the first 256 fg / bg indices
#define MAX_GT 512

typedef __attribute__((ext_vector_type(2))) float v2f;
typedef __attribute__((ext_vector_type(8))) float v8f;

__device__ __forceinline__ unsigned wang_hash(unsigned s) {
    s = (s ^ 61u) ^ (s >> 16);
    s *= 9u;
    s ^= s >> 4;
    s *= 0x27d4eb2du;
    s ^= s >> 15;
    return s;
}

// ---------------------------------------------------------------------------
// Kernel A: pairwise IoU + row max/argmax.
// One wave handles a 16-ROI tile against all GT boxes in 16-wide N tiles.
// The union's additive term area_a[m] + area_b[n] is computed as a rank-2
// outer product on the matrix unit via V_WMMA_F32_16X16X4_F32; intersections
// run branchless on the VALU using the WMMA C/D lane layout
// (M = v + 8*(lane>=16), N = lane & 15). Division is replaced by the fast
// TRANS-pipe reciprocal: union > 0 always, and inter==0 -> iou==0 exactly.
// ---------------------------------------------------------------------------
__global__ __launch_bounds__(256) void iou_argmax_wmma(
    const float* __restrict__ rois, int n_rois,
    const float* __restrict__ gt, int n_gt,
    float* __restrict__ max_ov, int* __restrict__ gt_as)
{
    __shared__ float sx1[MAX_GT], sy1[MAX_GT], sx2[MAX_GT], sy2[MAX_GT], sar[MAX_GT];

    int t = threadIdx.x;
    for (int j = t; j < n_gt; j += 256) {
        float x1 = gt[j * 5 + 0], y1 = gt[j * 5 + 1];
        float x2 = gt[j * 5 + 2], y2 = gt[j * 5 + 3];
        sx1[j] = x1; sy1[j] = y1; sx2[j] = x2; sy2[j] = y2;
        sar[j] = (x2 - x1 + 1.f) * (y2 - y1 + 1.f);
    }
    __syncthreads();

    int lane = t & 31;
    int wave = t >> 5;
    int roiBase = (blockIdx.x * 8 + wave) * 16;

    __builtin_prefetch(rois + (size_t)roiBase * 5, 0, 1);  // global_prefetch_b8

    // Float half-mask: 1.0 for lanes 0-15, 0.0 for lanes 16-31. One cndmask
    // at setup; all later uses are plain multiplies (no EXEC manipulation).
    float half0 = (lane < 16) ? 1.f : 0.f;

    // A operand (16x4 f32): lanes 0-15 carry (K0,K1) = (area_a[m], 1.0) for
    // row m = lane; lanes 16-31 carry (K2,K3) = (0,0).
    v2f amat;
    {
        int m = roiBase + (lane & 15);
        m = m < n_rois ? m : n_rois - 1;
        const float* rp = rois + (size_t)m * 5;
        float w = rp[3] - rp[1] + 1.f;
        float h = rp[4] - rp[2] + 1.f;
        amat.x = w * h * half0;
        amat.y = half0;
    }

    // Per-lane ROI coords keyed by the WMMA D layout: M = v + 8*(lane>=16).
    float ax1[8], ay1[8], ax2[8], ay2[8];
    int mBase = roiBase + ((lane >> 4) << 3);
    #pragma unroll
    for (int v = 0; v < 8; ++v) {
        int m = mBase + v;
        m = m < n_rois ? m : n_rois - 1;
        const float* rp = rois + (size_t)m * 5;
        ax1[v] = rp[1]; ay1[v] = rp[2]; ax2[v] = rp[3]; ay2[v] = rp[4];
    }

    float best[8];
    int   besti[8];
    #pragma unroll
    for (int v = 0; v < 8; ++v) { best[v] = -1.f; besti[v] = 0; }

    int n = lane & 15;
    for (int tile = 0; tile < n_gt; tile += 16) {
        int j = tile + n;
        // Unconditional LDS reads (broadcast-friendly); mask by multiply.
        float arj = sar[j];
        float bx1 = sx1[j], by1 = sy1[j], bx2 = sx2[j], by2 = sy2[j];

        // B operand (4x16 f32): lanes 0-15 carry (K0-row,K1-row) = (1, area_b[n]).
        v2f bmat;
        bmat.x = half0;
        bmat.y = arj * half0;
        v8f c = {};
        v8f d = __builtin_amdgcn_wmma_f32_16x16x4_f32(
            false, amat, false, bmat, (short)0, c, false, false);

        #pragma unroll
        for (int v = 0; v < 8; ++v) {
            float iw = fminf(ax2[v], bx2) - fmaxf(ax1[v], bx1) + 1.f;
            float ih = fminf(ay2[v], by2) - fmaxf(ay1[v], by1) + 1.f;
            iw = fmaxf(iw, 0.f);
            ih = fmaxf(ih, 0.f);
            float inter = iw * ih;
            float uni = d[v] - inter;              // area_a + area_b - inter (WMMA)
            // union > 0 always; inter == 0 gives iou == 0 exactly, matching
            // jnp.where(inter==0, 0, inter/union). Branchless fast rcp.
            float iou = inter * __builtin_amdgcn_rcpf(uni);
            bool gt_best = iou > best[v];
            best[v]  = gt_best ? iou : best[v];
            besti[v] = gt_best ? j   : besti[v];
        }
    }

    // Reduce (max, min-index-on-tie) across the 16 lanes that share M rows.
    #pragma unroll
    for (int v = 0; v < 8; ++v) {
        float bv = best[v];
        int   bi = besti[v];
        #pragma unroll
        for (int msk = 8; msk >= 1; msk >>= 1) {
            float ov = __shfl_xor(bv, msk, 16);
            int   oi = __shfl_xor(bi, msk, 16);
            if (ov > bv || (ov == bv && oi < bi)) { bv = ov; bi = oi; }
        }
        if (n == 0) {
            int r = mBase + v;
            if (r < n_rois) { max_ov[r] = bv; gt_as[r] = bi; }
        }
    }
}

// ---------------------------------------------------------------------------
// Kernel B1: per-block fg/bg counts (ballot + popcount).
// ---------------------------------------------------------------------------
__global__ __launch_bounds__(256) void count_kernel(
    const float* __restrict__ max_ov, int n,
    int* __restrict__ fgCnt, int* __restrict__ bgCnt)
{
    int i = blockIdx.x * 256 + threadIdx.x;
    int fg = 0, bg = 0;
    if (i < n) {
        float m = max_ov[i];
        fg = (m >= FG_THRESH);
        bg = (m < BG_HI) && (m >= BG_LO);
    }
    unsigned long long bf = __ballot(fg);
    unsigned long long bb = __ballot(bg);
    __shared__ int sf[8], sb[8];
    int lane = threadIdx.x & 31, wave = threadIdx.x >> 5;
    if (lane == 0) { sf[wave] = __popcll(bf); sb[wave] = __popcll(bb); }
    __syncthreads();
    if (threadIdx.x == 0) {
        int f = 0, b = 0;
        for (int w = 0; w < 8; ++w) { f += sf[w]; b += sb[w]; }
        fgCnt[blockIdx.x] = f;
        bgCnt[blockIdx.x] = b;
    }
}

// ---------------------------------------------------------------------------
// Kernel B2: single-block exclusive scan over block counts (nb <= 512).
// ---------------------------------------------------------------------------
__global__ __launch_bounds__(512) void scan_kernel(
    const int* __restrict__ fgCnt, const int* __restrict__ bgCnt, int nb,
    int* __restrict__ fgOff, int* __restrict__ bgOff, int* __restrict__ totals)
{
    __shared__ int f[512], b[512];
    int t = threadIdx.x;
    int fv0 = (t < nb) ? fgCnt[t] : 0;
    int bv0 = (t < nb) ? bgCnt[t] : 0;
    f[t] = fv0;
    b[t] = bv0;
    __syncthreads();
    for (int off = 1; off < 512; off <<= 1) {
        int fv = (t >= off) ? f[t - off] : 0;
        int bv = (t >= off) ? b[t - off] : 0;
        __syncthreads();
        f[t] += fv;
        b[t] += bv;
        __syncthreads();
    }
    if (t < nb) {
        fgOff[t] = f[t] - fv0;   // exclusive prefix
        bgOff[t] = b[t] - bv0;
    }
    if (t == 0) { totals[0] = f[511]; totals[1] = b[511]; }
}

// ---------------------------------------------------------------------------
// Kernel B3: rank-ordered scatter of the first CAP fg/bg indices.
// ---------------------------------------------------------------------------
__global__ __launch_bounds__(256) void scatter_kernel(
    const float* __restrict__ max_ov, int n,
    const int* __restrict__ fgOff, const int* __restrict__ bgOff,
    int* __restrict__ fgIdx, int* __restrict__ bgIdx)
{
    int i = blockIdx.x * 256 + threadIdx.x;
    int fg = 0, bg = 0;
    if (i < n) {
        float m = max_ov[i];
        fg = (m >= FG_THRESH);
        bg = (m < BG_HI) && (m >= BG_LO);
    }
    unsigned long long bf = __ballot(fg);
    unsigned long long bb = __ballot(bg);
    int lane = threadIdx.x & 31, wave = threadIdx.x >> 5;
    __shared__ int sf[8], sb[8], swf[8], swb[8];
    if (lane == 0) { sf[wave] = __popcll(bf); sb[wave] = __popcll(bb); }
    __syncthreads();
    if (threadIdx.x == 0) {
        int af = 0, ab = 0;
        for (int w = 0; w < 8; ++w) {
            swf[w] = af; af += sf[w];
            swb[w] = ab; ab += sb[w];
        }
    }
    __syncthreads();
    unsigned long long lmask = ((unsigned long long)1 << lane) - 1ull;
    if (fg) {
        int pos = fgOff[blockIdx.x] + swf[wave] + __popcll(bf & lmask);
        if (pos < CAP) fgIdx[pos] = i;
    }
    if (bg) {
        int pos = bgOff[blockIdx.x] + swb[wave] + __popcll(bb & lmask);
        if (pos < CAP) bgIdx[pos] = i;
    }
}

// ---------------------------------------------------------------------------
// Kernel C: sample 256 ROIs, bbox-transform, one-hot scatter of targets.
// Fully overwrites every output element (d_out is poisoned by the harness).
// ---------------------------------------------------------------------------
__global__ __launch_bounds__(256) void finalize_kernel(
    const float* __restrict__ rpn_rois, const float* __restrict__ rpn_scores,
    const float* __restrict__ gt_boxes,
    const float* __restrict__ max_ov, const int* __restrict__ gt_as,
    const int* __restrict__ fgIdx, const int* __restrict__ bgIdx,
    const int* __restrict__ totals, float* __restrict__ out)
{
    int t = threadIdx.x;
    int totFg = totals[0], totBg = totals[1];
    int fg_count = min(FG_ROIS, totFg);
    int bg_count = min(ROIS_PER_IMAGE - fg_count, totBg);
    int n_samp = fg_count + bg_count;
    int n_safe = max(n_samp, 1);

    int pos = t;
    if (pos >= n_samp) pos = (int)(wang_hash((unsigned)(t + 1013904223u)) % (unsigned)n_safe);
    int sel = 0;
    if (n_samp > 0)
        sel = (pos < fg_count) ? fgIdx[min(pos, CAP - 1)]
                               : bgIdx[min(pos - fg_count, CAP - 1)];

    int gtid = gt_as[sel];
    float label = gt_boxes[gtid * 5 + 4];
    float m = max_ov[sel];
    if (m < BG_HI && m >= BG_LO) label = 0.f;

    float r0  = rpn_rois[(size_t)sel * 5 + 0];
    float ex1 = rpn_rois[(size_t)sel * 5 + 1];
    float ey1 = rpn_rois[(size_t)sel * 5 + 2];
    float ex2 = rpn_rois[(size_t)sel * 5 + 3];
    float ey2 = rpn_rois[(size_t)sel * 5 + 4];
    float gx1 = gt_boxes[gtid * 5 + 0], gy1 = gt_boxes[gtid * 5 + 1];
    float gx2 = gt_boxes[gtid * 5 + 2], gy2 = gt_boxes[gtid * 5 + 3];

    float ew = ex2 - ex1 + 1.f, eh = ey2 - ey1 + 1.f;
    float ecx = ex1 + 0.5f * ew, ecy = ey1 + 0.5f * eh;
    float gw = gx2 - gx1 + 1.f, gh = gy2 - gy1 + 1.f;
    float gcx = gx1 + 0.5f * gw, gcy = gy1 + 0.5f * gh;
    float t0 = ((gcx - ecx) / ew) * 10.f;      // (x - mean)/0.1
    float t1 = ((gcy - ecy) / eh) * 10.f;
    float t2 = logf(gw / ew) * 5.f;            // /0.2
    float t3 = logf(gh / eh) * 5.f;

    const int W = 4 * NUM_CLASSES;
    float* o_rois = out;
    float* o_sc = out + ROIS_PER_IMAGE * 5;
    float* o_lb = o_sc + ROIS_PER_IMAGE;
    float* o_bt = o_lb + ROIS_PER_IMAGE;
    float* o_bi = o_bt + ROIS_PER_IMAGE * W;
    float* o_bo = o_bi + ROIS_PER_IMAGE * W;

    o_rois[t * 5 + 0] = r0;
    o_rois[t * 5 + 1] = ex1;
    o_rois[t * 5 + 2] = ey1;
    o_rois[t * 5 + 3] = ex2;
    o_rois[t * 5 + 4] = ey2;
    o_sc[t] = rpn_scores[sel];
    o_lb[t] = label;

    for (int k = 0; k < W; ++k) {
        o_bt[t * W + k] = 0.f;
        o_bi[t * W + k] = 0.f;
        o_bo[t * W + k] = 0.f;
    }
    if (label > 0.f) {
        int c0 = ((int)label) * 4;
        o_bt[t * W + c0 + 0] = t0;
        o_bt[t * W + c0 + 1] = t1;
        o_bt[t * W + c0 + 2] = t2;
        o_bt[t * W + c0 + 3] = t3;
        o_bi[t * W + c0 + 0] = 1.f; o_bi[t * W + c0 + 1] = 1.f;
        o_bi[t * W + c0 + 2] = 1.f; o_bi[t * W + c0 + 3] = 1.f;
        o_bo[t * W + c0 + 0] = 1.f; o_bo[t * W + c0 + 1] = 1.f;
        o_bo[t * W + c0 + 2] = 1.f; o_bo[t * W + c0 + 3] = 1.f;
    }
}

extern "C" void kernel_launch(void* const* d_in, const int* in_sizes, int n_in,
                              void* d_out, int out_size, void* d_ws, size_t ws_size,
                              hipStream_t stream)
{
    const float* rpn_rois   = (const float*)d_in[0];
    const float* rpn_scores = (const float*)d_in[1];
    const float* gt_boxes   = (const float*)d_in[2];
    int n_rois = in_sizes[0] / 5;
    int n_gt   = in_sizes[2] / 5;

    // Workspace carving (256B-aligned slices, < 1 MB total).
    char* ws = (char*)d_ws;
    size_t off = 0;
    auto alloc = [&](size_t bytes) -> void* {
        void* p = ws + off;
        off = (off + bytes + 255) & ~(size_t)255;
        return p;
    };
    float* max_ov = (float*)alloc((size_t)n_rois * 4);
    int*   gt_as  = (int*)alloc((size_t)n_rois * 4);
    int*   fgCnt  = (int*)alloc(512 * 4);
    int*   bgCnt  = (int*)alloc(512 * 4);
    int*   fgOff  = (int*)alloc(512 * 4);
    int*   bgOff  = (int*)alloc(512 * 4);
    int*   totals = (int*)alloc(2 * 4);
    int*   fgIdx  = (int*)alloc(CAP * 4);
    int*   bgIdx  = (int*)alloc(CAP * 4);

    int wavesTotal = (n_rois + 15) / 16;
    int blocksA = (wavesTotal + 7) / 8;
    iou_argmax_wmma<<<blocksA, 256, 0, stream>>>(rpn_rois, n_rois, gt_boxes, n_gt,
                                                 max_ov, gt_as);

    int nb = (n_rois + 255) / 256;   // 391 for N_ROIS = 100000 (<= 512)
    count_kernel<<<nb, 256, 0, stream>>>(max_ov, n_rois, fgCnt, bgCnt);
    scan_kernel<<<1, 512, 0, stream>>>(fgCnt, bgCnt, nb, fgOff, bgOff, totals);
    scatter_kernel<<<nb, 256, 0, stream>>>(max_ov, n_rois, fgOff, bgOff, fgIdx, bgIdx);
    finalize_kernel<<<1, 256, 0, stream>>>(rpn_rois, rpn_scores, gt_boxes,
                                           max_ov, gt_as, fgIdx, bgIdx, totals,
                                           (float*)d_out);
}